// KnowledgeRetriever_53042846106203
// MI455X (gfx1250) — compile-verified
//
#include <hip/hip_runtime.h>
#include <hip/hip_bf16.h>

// KnowledgeRetriever: reference output reduces to broadcast(mean_k(knowledge)).
//   query_embedding [4,1024,512] f32 : DEAD (argsort over all K=64 then mean is
//                                      permutation-invariant)
//   knowledge       [64,512]     f32 : only live input
//   out             [4,1024,512] f32 : every row = column-mean of knowledge
//
// Kernel 1: column mean via V_WMMA_F32_16X16X4_F32 (D = (1/64 * ones) x knowledge)
// Kernel 2: float4 broadcast store, HBM-write-bandwidth bound (~8 MB).

typedef __attribute__((ext_vector_type(2))) float v2f;
typedef __attribute__((ext_vector_type(8))) float v8f;

#define EMB 512
#define KCH 64
#define ROWS 4096   // 4 * 1024

// One block, 256 threads = 8 waves. Each wave owns 4 column-tiles of 16.
// For each tile: accumulate 16 chained WMMAs over K=64 (4 per step).
// A is uniformly 1/64 so D = column mean; all D rows identical.
__global__ void __launch_bounds__(256)
kr_mean_wmma(const float* __restrict__ knowledge, float* __restrict__ mean) {
    const int lane  = threadIdx.x & 31;
    const int wave  = threadIdx.x >> 5;     // 0..7
    const int col   = lane & 15;            // N within tile
    const int khalf = (lane >> 4) << 1;     // lanes 0-15 -> K pair {0,1}; 16-31 -> {2,3}

    v2f a;
    a.x = 1.0f / 64.0f;
    a.y = 1.0f / 64.0f;

    #pragma unroll
    for (int t = 0; t < 4; ++t) {
        const int n0 = (wave + 8 * t) * 16;     // 0..496
        v8f c = {};
        #pragma unroll
        for (int k0 = 0; k0 < KCH; k0 += 4) {
            v2f b;
            b.x = knowledge[(k0 + khalf + 0) * EMB + n0 + col];
            b.y = knowledge[(k0 + khalf + 1) * EMB + n0 + col];
            // D = A(16x4, all 1/64) * B(4x16 knowledge tile) + C
            c = __builtin_amdgcn_wmma_f32_16x16x4_f32(
                    /*neg_a=*/false, a, /*neg_b=*/false, b,
                    /*c_mod=*/(short)0, c, /*reuse_a=*/false, /*reuse_b=*/false);
        }
        // D row M=0 lives in VGPR c[0], lanes 0..15 hold N=0..15.
        if (lane < 16) mean[n0 + col] = c[0];
    }
}

// Broadcast mean (512 f32 = 128 float4, L1/L2 resident) to 4096 rows.
// 524288 float4 stores, fully coalesced -> store-bandwidth bound.
__global__ void __launch_bounds__(256)
kr_broadcast(const float4* __restrict__ mean4, float4* __restrict__ out4, int n4) {
    const int idx = blockIdx.x * blockDim.x + threadIdx.x;
    if (idx < n4) {
        out4[idx] = mean4[idx & (EMB / 4 - 1)];   // col4 = idx % 128
    }
}

extern "C" void kernel_launch(void* const* d_in, const int* in_sizes, int n_in,
                              void* d_out, int out_size, void* d_ws, size_t ws_size,
                              hipStream_t stream) {
    // d_in[0]: query_embedding (dead), d_in[1]: knowledge [64,512] f32
    const float* knowledge = (const float*)d_in[1];
    float* out  = (float*)d_out;
    float* mean = (float*)d_ws;            // 512 floats = 2 KB scratch

    kr_mean_wmma<<<1, 256, 0, stream>>>(knowledge, mean);

    const int n4 = ROWS * EMB / 4;         // 524288 float4
    kr_broadcast<<<(n4 + 255) / 256, 256, 0, stream>>>(
        (const float4*)mean, (float4*)out, n4);

    (void)d_in; (void)in_sizes; (void)n_in; (void)out_size; (void)ws_size;
}